// SpanPredictionModule_48318382080425
// MI455X (gfx1250) — compile-verified
//
#include <hip/hip_runtime.h>
#include <math.h>

typedef __attribute__((ext_vector_type(2))) float v2f;
typedef __attribute__((ext_vector_type(8))) float v8f;

#define L_DIM 2048
#define NT 256
#define EPT 8   // elements per thread: NT*EPT == L_DIM

__global__ __launch_bounds__(NT) void span_decode_kernel(
    const float* __restrict__ start_logits,
    const float* __restrict__ end_logits,
    float* __restrict__ out, int B)
{
    const int t = threadIdx.x;
    const int b = blockIdx.x;

    __shared__ float smx[NT];      // suffix-max scan of end chunks
    __shared__ int   smi[NT];      // argmax index for smx
    __shared__ float sms[NT];      // suffix-sum scan of exp(end - Emax) chunks
    __shared__ float smscore[NT];  // candidate score reduce
    __shared__ int   smbi[NT];
    __shared__ int   smbj[NT];
    __shared__ float smp[NT];      // per-thread denominator partials

    const float* erow = end_logits   + (size_t)b * L_DIM;
    const float* srow = start_logits + (size_t)b * L_DIM;

    // ---- load this thread's contiguous 8-element chunk (two b128 loads) ----
    float ev[EPT], sv[EPT];
    {
        const float4* e4 = reinterpret_cast<const float4*>(erow + t * EPT);
        const float4* s4 = reinterpret_cast<const float4*>(srow + t * EPT);
        float4 e0 = e4[0], e1 = e4[1];
        float4 s0 = s4[0], s1 = s4[1];
        ev[0]=e0.x; ev[1]=e0.y; ev[2]=e0.z; ev[3]=e0.w;
        ev[4]=e1.x; ev[5]=e1.y; ev[6]=e1.z; ev[7]=e1.w;
        sv[0]=s0.x; sv[1]=s0.y; sv[2]=s0.z; sv[3]=s0.w;
        sv[4]=s1.x; sv[5]=s1.y; sv[6]=s1.z; sv[7]=s1.w;
    }
    const float NEG_INF = -__builtin_inff();

    // ---- Phase 1: suffix max of end (with first-occurrence argmax) ----
    float lm[EPT]; int lj[EPT];
    lm[EPT-1] = ev[EPT-1]; lj[EPT-1] = t*EPT + (EPT-1);
    #pragma unroll
    for (int k = EPT-2; k >= 0; --k) {
        if (ev[k] >= lm[k+1]) { lm[k] = ev[k]; lj[k] = t*EPT + k; }
        else                  { lm[k] = lm[k+1]; lj[k] = lj[k+1]; }
    }
    smx[t] = lm[0]; smi[t] = lj[0];
    __syncthreads();
    // inclusive suffix scan over 256 chunk aggregates (left wins ties)
    for (int off = 1; off < NT; off <<= 1) {
        float vm = NEG_INF; int vi = 0;
        const bool has = (t + off) < NT;
        if (has) { vm = smx[t + off]; vi = smi[t + off]; }
        __syncthreads();
        if (has && vm > smx[t]) { smx[t] = vm; smi[t] = vi; }
        __syncthreads();
    }
    const float Emax = smx[0];
    const float exm  = (t < NT-1) ? smx[t+1] : NEG_INF;
    const int   exj  = (t < NT-1) ? smi[t+1] : 0;
    float m_[EPT]; int jm[EPT];
    #pragma unroll
    for (int k = 0; k < EPT; ++k) {
        if (lm[k] >= exm) { m_[k] = lm[k]; jm[k] = lj[k]; }
        else              { m_[k] = exm;   jm[k] = exj;   }
    }

    // ---- Phase 2: suffix sums T_i = sum_{j>=i} exp(e_j - Emax) ----
    float lsum[EPT];
    lsum[EPT-1] = __expf(ev[EPT-1] - Emax) * 0.0f + expf(ev[EPT-1] - Emax);
    #pragma unroll
    for (int k = EPT-2; k >= 0; --k) lsum[k] = expf(ev[k] - Emax) + lsum[k+1];
    sms[t] = lsum[0];
    __syncthreads();
    for (int off = 1; off < NT; off <<= 1) {
        float v = 0.0f;
        const bool has = (t + off) < NT;
        if (has) v = sms[t + off];
        __syncthreads();
        if (has) sms[t] += v;
        __syncthreads();
    }
    const float exs = (t < NT-1) ? sms[t+1] : 0.0f;
    float T[EPT];
    #pragma unroll
    for (int k = 0; k < EPT; ++k) T[k] = lsum[k] + exs;

    // ---- Phase 3: best span M = max_i (s_i + m_i), first-occurrence ties ----
    float bestS = NEG_INF; int bestI = 0, bestJ = 0;
    #pragma unroll
    for (int k = 0; k < EPT; ++k) {
        const float sc = sv[k] + m_[k];
        if (sc > bestS) { bestS = sc; bestI = t*EPT + k; bestJ = jm[k]; }
    }
    smscore[t] = bestS; smbi[t] = bestI; smbj[t] = bestJ;
    __syncthreads();
    for (int off = NT/2; off > 0; off >>= 1) {
        if (t < off) {
            const float o = smscore[t + off];
            if (o > smscore[t]) {
                smscore[t] = o; smbi[t] = smbi[t + off]; smbj[t] = smbj[t + off];
            }
        }
        __syncthreads();
    }
    const float M = smscore[0];

    // ---- Phase 4: denominator partials ----
    float pt = 0.0f;
    #pragma unroll
    for (int k = 0; k < EPT; ++k) pt += expf(sv[k] + Emax - M) * T[k];
    smp[t] = pt;
    __syncthreads();

    // ---- Phase 5: wave-0 reduces the 256 partials on the matrix pipe ----
    // Four chained V_WMMA_F32_16X16X4_F32 with B = ones: D[m,n] accumulates
    // row sums of four 16x4 tiles of smp (exact f32 math). Column total =
    // per-lane sum of 8 acc regs + partner lane (l ^ 16).
    if (t < 32) {
        const int lane = t;
        const int m    = lane & 15;
        const int koff = (lane >> 4) ? 2 : 0;   // A VGPR0: K=0 / K=2 ; VGPR1: K=1 / K=3
        v2f bones; bones.x = 1.0f; bones.y = 1.0f;
        v8f acc = {0.f, 0.f, 0.f, 0.f, 0.f, 0.f, 0.f, 0.f};
        #pragma unroll
        for (int c = 0; c < 4; ++c) {
            v2f a;
            a.x = smp[c * 64 + m * 4 + koff + 0];
            a.y = smp[c * 64 + m * 4 + koff + 1];
            acc = __builtin_amdgcn_wmma_f32_16x16x4_f32(
                /*neg_a=*/false, a, /*neg_b=*/false, bones,
                /*c_mod=*/(short)0, acc, /*reuse_a=*/false, /*reuse_b=*/false);
        }
        float rs = acc[0]+acc[1]+acc[2]+acc[3]+acc[4]+acc[5]+acc[6]+acc[7];
        rs += __shfl_xor(rs, 16, 32);           // add rows M=8..15 (partner lane)
        if (lane == 0) {
            const float denom = rs;             // sum of all 256 partials
            out[b]         = 1.0f / denom;      // exp(M-M)/denom
            out[B + b]     = (float)smbi[0];    // span_start_indices
            out[2*B + b]   = (float)smbj[0];    // span_end_indices
        }
    }
}

extern "C" void kernel_launch(void* const* d_in, const int* in_sizes, int n_in,
                              void* d_out, int out_size, void* d_ws, size_t ws_size,
                              hipStream_t stream) {
    (void)n_in; (void)out_size; (void)d_ws; (void)ws_size;
    const float* s = (const float*)d_in[0];
    const float* e = (const float*)d_in[1];
    float* out = (float*)d_out;
    const int B = in_sizes[0] / L_DIM;
    span_decode_kernel<<<dim3(B), dim3(NT), 0, stream>>>(s, e, out, B);
}